// FDSE_GraphSAGE_Layer_61443802137343
// MI455X (gfx1250) — compile-verified
//
#include <hip/hip_runtime.h>

typedef __attribute__((ext_vector_type(2))) float v2f;
typedef __attribute__((ext_vector_type(8))) float v8f;

constexpr int N_NODES = 100000;
constexpr int N_EDGES = 3200000;
constexpr int D       = 128;      // D_IN == D_OUT
constexpr float BN_EPS = 1e-5f;

// ---------------------------------------------------------------------------
// 0) zero workspace (agg_sum, deg, sum, sumsq)
// ---------------------------------------------------------------------------
__global__ void k_zero(float* __restrict__ p, long n) {
  long i = (long)blockIdx.x * blockDim.x + threadIdx.x;
  long stride = (long)gridDim.x * blockDim.x;
  for (; i < n; i += stride) p[i] = 0.0f;
}

// ---------------------------------------------------------------------------
// 1) edge scatter: one wave32 per edge; each lane gathers float4 of x[src]
//    and atomically adds into agg[dst]; lane 0 bumps the degree counter.
// ---------------------------------------------------------------------------
__global__ void k_scatter(const float* __restrict__ x,
                          const int* __restrict__ src,
                          const int* __restrict__ dst,
                          float* __restrict__ agg,
                          float* __restrict__ deg) {
  const int lane = threadIdx.x & 31;
  long wave   = ((long)blockIdx.x * blockDim.x + threadIdx.x) >> 5;
  long nwaves = ((long)gridDim.x * blockDim.x) >> 5;
  for (long e = wave; e < N_EDGES; e += nwaves) {
    const int s = src[e];
    const int d = dst[e];
    const float4 v = ((const float4*)(x + (long)s * D))[lane];
    float* ad = agg + (long)d * D + lane * 4;
    atomicAdd(ad + 0, v.x);
    atomicAdd(ad + 1, v.y);
    atomicAdd(ad + 2, v.z);
    atomicAdd(ad + 3, v.w);
    if (lane == 0) atomicAdd(deg + d, 1.0f);
  }
}

// ---------------------------------------------------------------------------
// 2) fused SAGE GEMM:  x_raw = (agg/deg) @ W_l + b_l + x @ W_r
//    block = 256 threads = 8 waves; block covers 16 rows x 128 cols.
//    Each wave computes one 16x16 tile with V_WMMA_F32_16X16X4_F32,
//    accumulating both GEMMs into the same v8f accumulator.
// ---------------------------------------------------------------------------
__global__ void __launch_bounds__(256)
k_sage_gemm(const float* __restrict__ x, const float* __restrict__ agg,
            const float* __restrict__ deg, const float* __restrict__ Wl,
            const float* __restrict__ bl, const float* __restrict__ Wr,
            float* __restrict__ xraw) {
  constexpr int LDSP = 132;                 // pad 128 -> 132 (bank-conflict free)
  __shared__ float As[16][LDSP];            // degree-normalized aggregate rows
  __shared__ float Xs[16][LDSP];            // self-feature rows
  const int row0 = blockIdx.x * 16;

  // stage 16x128 panel of both A operands into LDS (uniform loop, EXEC full)
  for (int i = threadIdx.x; i < 16 * D; i += 256) {
    const int r = i >> 7, c = i & (D - 1);
    const long g = (long)(row0 + r) * D + c;
    const float dg = fmaxf(deg[row0 + r], 1.0f);
    As[r][c] = agg[g] / dg;
    Xs[r][c] = x[g];
  }
  __syncthreads();

  const int wave = threadIdx.x >> 5;
  const int lane = threadIdx.x & 31;
  const int col0 = wave * 16;               // 8 waves cover 128 output columns
  const int mr   = lane & 15;               // A-row (lanes 0-15 / 16-31) and B/D column
  const int kb   = (lane >> 4) * 2;         // K sub-offset for upper half-wave

  v8f acc = {};
  // GEMM 1: (agg/deg) @ W_l
  #pragma unroll 4
  for (int k = 0; k < D; k += 4) {
    v2f a, b;
    a.x = As[mr][k + kb];
    a.y = As[mr][k + kb + 1];
    b.x = Wl[(k + kb) * D + col0 + mr];
    b.y = Wl[(k + kb + 1) * D + col0 + mr];
    acc = __builtin_amdgcn_wmma_f32_16x16x4_f32(false, a, false, b,
                                                (short)0, acc, false, false);
  }
  // GEMM 2: x @ W_r  (same accumulator)
  #pragma unroll 4
  for (int k = 0; k < D; k += 4) {
    v2f a, b;
    a.x = Xs[mr][k + kb];
    a.y = Xs[mr][k + kb + 1];
    b.x = Wr[(k + kb) * D + col0 + mr];
    b.y = Wr[(k + kb + 1) * D + col0 + mr];
    acc = __builtin_amdgcn_wmma_f32_16x16x4_f32(false, a, false, b,
                                                (short)0, acc, false, false);
  }

  // D layout: VGPR i -> row = i + 8*(lane>=16), col = lane&15
  const float bias  = bl[col0 + mr];
  const int   rbase = row0 + ((lane >> 4) * 8);
  #pragma unroll
  for (int i = 0; i < 8; ++i) {
    xraw[(long)(rbase + i) * D + col0 + mr] = acc[i] + bias;
  }
}

// ---------------------------------------------------------------------------
// 3) BN column stats: thread t owns column t; blocks stride over rows,
//    partial sums merged with global float atomics.
// ---------------------------------------------------------------------------
__global__ void k_stats(const float* __restrict__ xraw,
                        float* __restrict__ sum, float* __restrict__ sumsq) {
  const int c = threadIdx.x;                // 0..127
  float s = 0.0f, s2 = 0.0f;
  for (int r = blockIdx.x; r < N_NODES; r += gridDim.x) {
    const float v = xraw[(long)r * D + c];
    s  += v;
    s2 += v * v;
  }
  atomicAdd(&sum[c], s);
  atomicAdd(&sumsq[c], s2);
}

// ---------------------------------------------------------------------------
// 4) finalize BN: scale = gamma*rsqrt(var+eps), shift = beta - mu*scale
// ---------------------------------------------------------------------------
__global__ void k_finalize(const float* __restrict__ sum,
                           const float* __restrict__ sumsq,
                           const float* __restrict__ gamma,
                           const float* __restrict__ beta,
                           float* __restrict__ scale,
                           float* __restrict__ shift) {
  const int c = threadIdx.x;
  const float inv_n = 1.0f / (float)N_NODES;
  const float mu  = sum[c] * inv_n;
  const float var = sumsq[c] * inv_n - mu * mu;
  const float sc  = gamma[c] * rsqrtf(var + BN_EPS);
  scale[c] = sc;
  shift[c] = beta[c] - mu * sc;
}

// ---------------------------------------------------------------------------
// 5) normalize pass: x_deskewed = x_raw*scale[c] + shift[c]
// ---------------------------------------------------------------------------
__global__ void k_norm(const float* __restrict__ xraw,
                       const float* __restrict__ scale,
                       const float* __restrict__ shift,
                       float* __restrict__ outd) {
  long i = (long)blockIdx.x * blockDim.x + threadIdx.x;
  const long n = (long)N_NODES * D;
  const long stride = (long)gridDim.x * blockDim.x;
  for (; i < n; i += stride) {
    const int c = (int)(i & (D - 1));
    outd[i] = xraw[i] * scale[c] + shift[c];
  }
}

// ---------------------------------------------------------------------------
extern "C" void kernel_launch(void* const* d_in, const int* in_sizes, int n_in,
                              void* d_out, int out_size, void* d_ws, size_t ws_size,
                              hipStream_t stream) {
  const float* x     = (const float*)d_in[0];
  const int*   edge  = (const int*)d_in[1];   // [2, N_EDGES]: row0=src, row1=dst
  const float* Wl    = (const float*)d_in[2];
  const float* bl    = (const float*)d_in[3];
  const float* Wr    = (const float*)d_in[4];
  const float* gamma = (const float*)d_in[5];
  const float* beta  = (const float*)d_in[6];

  float* xraw = (float*)d_out;                      // output 0: x_raw
  float* xdsk = (float*)d_out + (long)N_NODES * D;  // output 1: x_deskewed

  // workspace layout
  float* agg   = (float*)d_ws;                      // N_NODES * D
  float* deg   = agg + (long)N_NODES * D;           // N_NODES
  float* sum   = deg + N_NODES;                     // D
  float* sumsq = sum + D;                           // D
  float* scale = sumsq + D;                         // D
  float* shift = scale + D;                         // D

  const long nzero = (long)N_NODES * D + N_NODES + 2L * D;
  k_zero<<<2048, 256, 0, stream>>>(agg, nzero);
  k_scatter<<<4096, 256, 0, stream>>>(x, edge, edge + N_EDGES, agg, deg);
  k_sage_gemm<<<N_NODES / 16, 256, 0, stream>>>(x, agg, deg, Wl, bl, Wr, xraw);
  k_stats<<<512, 128, 0, stream>>>(xraw, sum, sumsq);
  k_finalize<<<1, 128, 0, stream>>>(sum, sumsq, gamma, beta, scale, shift);
  k_norm<<<2048, 256, 0, stream>>>(xraw, scale, shift, xdsk);
}